// EmbedRelationGATEncoder_90546500534896
// MI455X (gfx1250) — compile-verified
//
#include <hip/hip_runtime.h>
#include <math.h>

typedef float v2f __attribute__((ext_vector_type(2)));
typedef float v8f __attribute__((ext_vector_type(8)));

#define NEG_SLOPE 0.2f
#define LN_EPS 1e-5f

// ---------------- utility: float atomic max via ordered-int trick ----------
__device__ __forceinline__ void atomicMaxF(float* addr, float val) {
  unsigned u = __float_as_uint(val);
  if (u >> 31) {
    // negative float: more negative => larger unsigned; min-unsigned keeps max float
    atomicMin((unsigned int*)addr, u);
  } else {
    // non-negative float: int compare is monotone; any stored negative pattern is INT-negative
    atomicMax((int*)addr, (int)u);
  }
}

// ---------------- fill -----------------------------------------------------
__global__ void fill_f32(float* __restrict__ p, float v, size_t n) {
  size_t i = (size_t)blockIdx.x * blockDim.x + threadIdx.x;
  size_t stride = (size_t)gridDim.x * blockDim.x;
  for (; i < n; i += stride) p[i] = v;
}

// ---------------- f32 WMMA GEMM: C = A@B (+C if ACC) ----------------------
// A: [M,K] row-major, B: [K,Nc] row-major, C: [M,Nc] row-major.
// grid = (Nc/128, M/16), block = 256 (8 waves). Each wave owns a 16x16 tile.
// Compile-time K (% 4 == 0); Nc % 128 == 0, M % 16 == 0. EXEC all-ones.
// LDS row stride padded to K+2 (even -> 8B-aligned v2f ds loads, no 64-bank
// conflicts: lane l hits bank (2*l + kb) % 64).
template <int K, bool ACC>
__global__ __launch_bounds__(256)
void wmma_gemm_f32(const float* __restrict__ A, const float* __restrict__ B,
                   float* __restrict__ C, int Ncols) {
  constexpr int KP = K + 2;  // padded LDS row stride (floats)
  __shared__ float As[16 * KP];
  const int row0 = blockIdx.y * 16;
  const int tid = threadIdx.x;

  // cooperative, coalesced load of the 16 x K A tile (shared by all 8 waves)
#pragma unroll
  for (int i = tid; i < 16 * K; i += 256) {
    int r = i >> __builtin_ctz(K);   // i / K  (K is a power-of-two here)
    int c = i & (K - 1);             // i % K
    As[r * KP + c] = A[(size_t)(row0 + r) * K + c];
  }
  __syncthreads();

  const int lane = tid & 31;
  const int wave = tid >> 5;
  const int halfsel = lane >> 4;  // 0: lanes 0-15, 1: lanes 16-31
  const int col = blockIdx.x * 128 + wave * 16 + (lane & 15);
  const float* asrc = &As[(lane & 15) * KP + (halfsel << 1)];

  v8f c = {};
  if (ACC) {
#pragma unroll
    for (int v = 0; v < 8; ++v) {
      int r = v + (halfsel << 3);
      c[v] = C[(size_t)(row0 + r) * Ncols + col];
    }
  }

#pragma unroll 8
  for (int k = 0; k < K; k += 4) {
    int kb = k + (halfsel << 1);
    // A fragment (16x4): lane holds row = lane&15, K = kb, kb+1
    v2f a = *(const v2f*)(asrc + k);
    // B fragment (4x16): lane holds col, K = kb, kb+1
    v2f b;
    b.x = B[(size_t)kb * Ncols + col];
    b.y = B[(size_t)(kb + 1) * Ncols + col];
    c = __builtin_amdgcn_wmma_f32_16x16x4_f32(false, a, false, b, (short)0, c,
                                              false, false);
  }

#pragma unroll
  for (int v = 0; v < 8; ++v) {
    int r = v + (halfsel << 3);
    C[(size_t)(row0 + r) * Ncols + col] = c[v];
  }
}

// ---------------- attention projections: a = sum(h * att, axis=-1) --------
// One wave per (node, head). Hm: [Nn, H*C]; attS/attD: [H, C]; out: [Nn*H].
__global__ void att_dots(const float* __restrict__ Hm, const float* __restrict__ attS,
                         const float* __restrict__ attD, float* __restrict__ aS,
                         float* __restrict__ aD, int Nn, int H, int C) {
  int w = (int)(((size_t)blockIdx.x * blockDim.x + threadIdx.x) >> 5);
  int lane = threadIdx.x & 31;
  if (w >= Nn * H) return;
  int n = w / H;
  int h = w - n * H;
  const float* hp = Hm + (size_t)n * H * C + (size_t)h * C;
  float s = 0.f, d = 0.f;
  for (int cc = lane; cc < C; cc += 32) {
    float x = hp[cc];
    s += x * attS[h * C + cc];
    d += x * attD[h * C + cc];
  }
#pragma unroll
  for (int o = 16; o > 0; o >>= 1) {
    s += __shfl_down(s, o, 32);
    d += __shfl_down(d, o, 32);
  }
  if (lane == 0) { aS[w] = s; aD[w] = d; }
}

// ---------------- edge pass 1: leaky-relu score + segment max -------------
__global__ void edge_score_max(const long long* __restrict__ ei,
                               const float* __restrict__ aS, const float* __restrict__ aD,
                               float* __restrict__ esc, float* __restrict__ m,
                               long long E, long long Etot, int H) {
  long long e = (long long)blockIdx.x * blockDim.x + threadIdx.x;
  if (e >= Etot) return;
  long long s, d;
  if (e < E) { s = ei[e]; d = ei[E + e]; } else { s = d = e - E; }
  for (int h = 0; h < H; ++h) {
    float v = aS[s * H + h] + aD[d * H + h];
    v = v > 0.f ? v : NEG_SLOPE * v;
    esc[e * H + h] = v;
    atomicMaxF(&m[d * H + h], v);
  }
}

// ---------------- edge pass 2: exp(e - max) + segment sum -----------------
__global__ void edge_exp_sum(const long long* __restrict__ ei, float* __restrict__ esc,
                             const float* __restrict__ m, float* __restrict__ ssum,
                             long long E, long long Etot, int H) {
  long long e = (long long)blockIdx.x * blockDim.x + threadIdx.x;
  if (e >= Etot) return;
  long long d;
  if (e < E) { d = ei[E + e]; } else { d = e - E; }
  for (int h = 0; h < H; ++h) {
    float ex = expf(esc[e * H + h] - m[d * H + h]);
    esc[e * H + h] = ex;
    atomicAdd(&ssum[d * H + h], ex);
  }
}

// ---------------- edge pass 3: alpha-weighted message scatter -------------
// One wave per edge; lanes sweep the H*C feature vector (coalesced).
__global__ void edge_scatter(const long long* __restrict__ ei, const float* __restrict__ hsrc,
                             const float* __restrict__ esc, const float* __restrict__ ssum,
                             float* __restrict__ acc, long long E, long long Etot,
                             int H, int C) {
  long long w = ((long long)blockIdx.x * blockDim.x + threadIdx.x) >> 5;
  int lane = threadIdx.x & 31;
  if (w >= Etot) return;
  long long s, d;
  if (w < E) { s = ei[w]; d = ei[E + w]; } else { s = d = w - E; }
  int F = H * C;
  const float* hp = hsrc + (size_t)s * F;
  float* ap = acc + (size_t)d * F;
  for (int i = lane; i < F; i += 32) {
    int hh = i / C;
    float alpha = esc[w * H + hh] / ssum[d * H + hh];
    atomicAdd(&ap[i], hp[i] * alpha);
  }
}

// ---------------- bias + ELU (in place); F must be a power of two ---------
__global__ void bias_elu(float* __restrict__ x, const float* __restrict__ b,
                         size_t total, int F) {
  size_t i = (size_t)blockIdx.x * blockDim.x + threadIdx.x;
  size_t stride = (size_t)gridDim.x * blockDim.x;
  for (; i < total; i += stride) {
    float v = x[i] + b[(int)(i & (size_t)(F - 1))];
    x[i] = v > 0.f ? v : expm1f(v);
  }
}

// ---------------- final: y = acc2 + b2 + br; LayerNorm(gamma,beta) --------
// One block (256 threads) per node; C == 256.
__global__ __launch_bounds__(256)
void ln_out(const float* __restrict__ y0, const float* __restrict__ b2,
            const float* __restrict__ br, const float* __restrict__ g,
            const float* __restrict__ be, float* __restrict__ out, int C) {
  int n = blockIdx.x;
  int t = threadIdx.x;
  float v = y0[(size_t)n * C + t] + b2[t] + br[t];
  float s = v, q = v * v;
#pragma unroll
  for (int o = 16; o > 0; o >>= 1) {
    s += __shfl_down(s, o, 32);
    q += __shfl_down(q, o, 32);
  }
  __shared__ float ss[8], qq[8], stats[2];
  int w = t >> 5, l = t & 31;
  if (l == 0) { ss[w] = s; qq[w] = q; }
  __syncthreads();
  if (t == 0) {
    float S = 0.f, Q = 0.f;
#pragma unroll
    for (int i = 0; i < 8; ++i) { S += ss[i]; Q += qq[i]; }
    float mu = S / (float)C;
    stats[0] = mu;
    stats[1] = rsqrtf(Q / (float)C - mu * mu + LN_EPS);
  }
  __syncthreads();
  out[(size_t)n * C + t] = (v - stats[0]) * stats[1] * g[t] + be[t];
}

// ---------------------------------------------------------------------------
extern "C" void kernel_launch(void* const* d_in, const int* in_sizes, int n_in,
                              void* d_out, int out_size, void* d_ws, size_t ws_size,
                              hipStream_t stream) {
  const long long* edge_index = (const long long*)d_in[0];  // [2,E] int64
  // d_in[1] edge_type, d_in[3] rel_emb: unused downstream in the reference
  const float* x0       = (const float*)d_in[2];   // [N,128]
  const float* W1       = (const float*)d_in[4];   // [128,512]
  const float* att_src1 = (const float*)d_in[5];   // [4,128]
  const float* att_dst1 = (const float*)d_in[6];   // [4,128]
  const float* b1       = (const float*)d_in[7];   // [512]
  const float* W2       = (const float*)d_in[8];   // [512,256]
  const float* att_src2 = (const float*)d_in[9];   // [1,256]
  const float* att_dst2 = (const float*)d_in[10];  // [1,256]
  const float* b2       = (const float*)d_in[11];  // [256]
  const float* Wr       = (const float*)d_in[12];  // [128,256]
  const float* br       = (const float*)d_in[13];  // [256]
  const float* gamma    = (const float*)d_in[14];  // [256]
  const float* beta     = (const float*)d_in[15];  // [256]
  float* out = (float*)d_out;

  const long long E = in_sizes[0] / 2;     // 400000
  const int N = in_sizes[2] / 128;         // 50000 (multiple of 16)
  const long long Etot = E + N;            // with self loops
  const int H1 = 4, C1 = 128, F1 = 512, C2 = 256;

  // workspace layout (floats)
  float* ws = (float*)d_ws;
  size_t off = 0;
  float* h1   = ws + off; off += (size_t)N * F1;
  float* acc1 = ws + off; off += (size_t)N * F1;
  float* h2   = ws + off; off += (size_t)N * C2;
  float* acc2 = ws + off; off += (size_t)N * C2;
  float* esc1 = ws + off; off += (size_t)Etot * H1;
  float* esc2 = ws + off; off += (size_t)Etot;
  float* aS1  = ws + off; off += (size_t)N * H1;
  float* aD1  = ws + off; off += (size_t)N * H1;
  float* m1   = ws + off; off += (size_t)N * H1;
  float* s1   = ws + off; off += (size_t)N * H1;
  float* aS2  = ws + off; off += (size_t)N;
  float* aD2  = ws + off; off += (size_t)N;
  float* m2   = ws + off; off += (size_t)N;
  float* s2   = ws + off; off += (size_t)N;
  (void)off; (void)ws_size; (void)n_in; (void)out_size;

  auto cdiv = [](long long a, long long b) { return (int)((a + b - 1) / b); };

  // init accumulators / softmax state
  fill_f32<<<2048, 256, 0, stream>>>(acc1, 0.f, (size_t)N * F1);
  fill_f32<<<2048, 256, 0, stream>>>(acc2, 0.f, (size_t)N * C2);
  fill_f32<<<512, 256, 0, stream>>>(m1, -INFINITY, (size_t)N * H1);
  fill_f32<<<512, 256, 0, stream>>>(s1, 0.f, (size_t)N * H1);
  fill_f32<<<256, 256, 0, stream>>>(m2, -INFINITY, (size_t)N);
  fill_f32<<<256, 256, 0, stream>>>(s2, 0.f, (size_t)N);

  // ---- GAT layer 1 ----
  wmma_gemm_f32<128, false><<<dim3(F1 / 128, N / 16), 256, 0, stream>>>(x0, W1, h1, F1);
  att_dots<<<cdiv((long long)N * H1 * 32, 256), 256, 0, stream>>>(
      h1, att_src1, att_dst1, aS1, aD1, N, H1, C1);
  edge_score_max<<<cdiv(Etot, 256), 256, 0, stream>>>(edge_index, aS1, aD1, esc1, m1, E, Etot, H1);
  edge_exp_sum<<<cdiv(Etot, 256), 256, 0, stream>>>(edge_index, esc1, m1, s1, E, Etot, H1);
  edge_scatter<<<cdiv(Etot * 32, 256), 256, 0, stream>>>(edge_index, h1, esc1, s1, acc1,
                                                         E, Etot, H1, C1);
  bias_elu<<<2048, 256, 0, stream>>>(acc1, b1, (size_t)N * F1, F1);  // acc1 -> x1

  // ---- GAT layer 2 (1 head, mean over 1 head == identity) ----
  wmma_gemm_f32<512, false><<<dim3(C2 / 128, N / 16), 256, 0, stream>>>(acc1, W2, h2, C2);
  att_dots<<<cdiv((long long)N * 32, 256), 256, 0, stream>>>(
      h2, att_src2, att_dst2, aS2, aD2, N, 1, C2);
  edge_score_max<<<cdiv(Etot, 256), 256, 0, stream>>>(edge_index, aS2, aD2, esc2, m2, E, Etot, 1);
  edge_exp_sum<<<cdiv(Etot, 256), 256, 0, stream>>>(edge_index, esc2, m2, s2, E, Etot, 1);
  edge_scatter<<<cdiv(Etot * 32, 256), 256, 0, stream>>>(edge_index, h2, esc2, s2, acc2,
                                                         E, Etot, 1, C2);

  // ---- residual: acc2 += x0 @ Wr (accumulate mode) ----
  wmma_gemm_f32<128, true><<<dim3(C2 / 128, N / 16), 256, 0, stream>>>(x0, Wr, acc2, C2);

  // ---- y = acc2 + b2 + br; LayerNorm ----
  ln_out<<<N, 256, 0, stream>>>(acc2, b2, br, gamma, beta, out, C2);
}